// MyModel_87522843560254
// MI455X (gfx1250) — compile-verified
//
#include <hip/hip_runtime.h>
#include <math.h>

// GRU fused scan for MI455X (gfx1250, wave32).
// Each wave owns a 16-row batch tile and runs the full T=512 scan:
//   x_proj tile via v_wmma_f32_16x16x4_f32 (K=64 -> 16 k-steps, N=30 -> 2 tiles)
//   rec     tile via v_wmma_f32_16x16x4_f32 (K=10 padded to 12 -> 3 k-steps)
//   gates relayout through wave-private LDS (same-wave DS ops are in-order).
// Input tiles are prefetched with global_load_async_to_lds_b128 (double buffer,
// issue-then-wait<=8 so the next tile's DMAs stay in flight during the wait).
// Gate nonlinearities use the CDNA5 hardware v_tanh_f32 when available.

typedef __attribute__((ext_vector_type(2))) float v2f;
typedef __attribute__((ext_vector_type(8))) float v8f;

#define UNITS 10
#define N3    30        // 3*UNITS
#define BATCH 4096
#define TLEN  512
#define DIM   64

#define WAVES_PER_BLOCK 4
#define ROWS_PER_WAVE   16
#define TILE_STRIDE     68   // 64 + 4 pad floats: kills LDS bank conflicts on A-frag reads
#define TILE_FLOATS     (ROWS_PER_WAVE * TILE_STRIDE)   // 1088
#define STAGE_FLOATS    (16 * 32)                       // 512
#define H_STRIDE        12                              // 10 + 2 pad (K padded to 12)
#define H_FLOATS        (ROWS_PER_WAVE * H_STRIDE)      // 192
#define WAVE_LDS_FLOATS (2 * TILE_FLOATS + 2 * STAGE_FLOATS + H_FLOATS) // 3392

__device__ __forceinline__ v8f wmma_k4(v2f a, v2f b, v8f c) {
  // D = A(16x4,f32) * B(4x16,f32) + C(16x16,f32)
  return __builtin_amdgcn_wmma_f32_16x16x4_f32(false, a, false, b,
                                               (short)0, c, false, false);
}

// CDNA5 has a hardware transcendental tanh (v_tanh_f32).
__device__ __forceinline__ float fast_tanh(float x) {
#if __has_builtin(__builtin_amdgcn_tanhf)
  return __builtin_amdgcn_tanhf(x);
#else
  // tanh(x) = 2/(1+exp(-2x)) - 1, via v_exp_f32 + v_rcp_f32
  return __builtin_fmaf(2.0f, __builtin_amdgcn_rcpf(1.0f + __expf(-2.0f * x)), -1.0f);
#endif
}

__device__ __forceinline__ float fast_sigmoid(float x) {
#if __has_builtin(__builtin_amdgcn_tanhf)
  // sigmoid(x) = 0.5*tanh(x/2) + 0.5  -> one v_tanh_f32 + mul + fma
  return __builtin_fmaf(0.5f, __builtin_amdgcn_tanhf(0.5f * x), 0.5f);
#else
  return __builtin_amdgcn_rcpf(1.0f + __expf(-x));
#endif
}

__global__ __launch_bounds__(WAVES_PER_BLOCK * 32)
void gru_fused_kernel(const float* __restrict__ inputs,   // (B, T, D)
                      const float* __restrict__ wk,       // (D, 30)
                      const float* __restrict__ rk,       // (U, 30)
                      const float* __restrict__ bias,     // (2, 30)
                      float* __restrict__ out)            // (B, U)
{
  __shared__ float lds[WAVES_PER_BLOCK * WAVE_LDS_FLOATS];

  const int lane = threadIdx.x & 31;
  const int wave = threadIdx.x >> 5;
  const int row0 = (blockIdx.x * WAVES_PER_BLOCK + wave) * ROWS_PER_WAVE;

  float* wls   = &lds[wave * WAVE_LDS_FLOATS];
  float* tbuf0 = wls;                                 // input tile double buffer
  float* tbuf1 = wls + TILE_FLOATS;
  float* Sx    = wls + 2 * TILE_FLOATS;               // x_proj staging (16x32)
  float* Sr    = Sx + STAGE_FLOATS;                   // rec    staging (16x32)
  float* Hs    = Sr + STAGE_FLOATS;                   // h state (16x12), cols 10,11 = 0

  // LDS byte addresses for the async DMA destinations (static LDS, low 32 bits
  // of the flat address are the LDS byte offset).
  const unsigned tb0 = (unsigned)(uintptr_t)tbuf0;
  const unsigned tb1 = (unsigned)(uintptr_t)tbuf1;

  // h = 0 (including the 2 pad columns, which must stay 0 forever)
  for (int i = lane; i < H_FLOATS; i += 32) Hs[i] = 0.0f;

  const int half = (lane < 16) ? 0 : 1;  // A/B frag half: lanes16-31 carry K+2
  const int nlo  = lane & 15;

  // ---- preload B fragments: input kernel (64x30, N padded to 32) ----
  // B 4x16 layout: lanes 0-15 hold (K=4k+0 in v0, K=4k+1 in v1) for N=lane,
  //                lanes16-31 hold (K=4k+2, K=4k+3) for N=lane-16.
  v2f bx0[16], bx1[16];
#pragma unroll
  for (int k = 0; k < 16; ++k) {
    int r  = 4 * k + 2 * half;
    int n1 = 16 + nlo;
    float a0 = wk[r * N3 + nlo];
    float a1 = wk[(r + 1) * N3 + nlo];
    float b0 = (n1 < N3) ? wk[r * N3 + n1] : 0.0f;
    float b1 = (n1 < N3) ? wk[(r + 1) * N3 + n1] : 0.0f;
    v2f t0; t0.x = a0; t0.y = a1; bx0[k] = t0;
    v2f t1; t1.x = b0; t1.y = b1; bx1[k] = t1;
  }
  // ---- preload B fragments: recurrent kernel (10x30 -> K padded to 12) ----
  v2f br0[3], br1[3];
#pragma unroll
  for (int kk = 0; kk < 3; ++kk) {
    int r  = 4 * kk + 2 * half;
    int n1 = 16 + nlo;
    float a0 = (r     < UNITS) ? rk[r * N3 + nlo] : 0.0f;
    float a1 = (r + 1 < UNITS) ? rk[(r + 1) * N3 + nlo] : 0.0f;
    float b0 = (r     < UNITS && n1 < N3) ? rk[r * N3 + n1] : 0.0f;
    float b1 = (r + 1 < UNITS && n1 < N3) ? rk[(r + 1) * N3 + n1] : 0.0f;
    v2f t0; t0.x = a0; t0.y = a1; br0[kk] = t0;
    v2f t1; t1.x = b0; t1.y = b1; br1[kk] = t1;
  }
  // biases for this lane's output columns (folded into the accumulator init)
  const float bin0 = bias[nlo];
  const float bin1 = (16 + nlo < N3) ? bias[16 + nlo] : 0.0f;
  const float brc0 = bias[N3 + nlo];
  const float brc1 = (16 + nlo < N3) ? bias[N3 + 16 + nlo] : 0.0f;

  // ---- async prefetch of one 16x64 f32 input tile (4KB) into LDS ----
  // 256 x 16B chunks -> 8 b128 DMAs per lane, fully coalesced per row.
  auto prefetch = [&](int t, unsigned ldsb) {
#pragma unroll
    for (int i = 0; i < 8; ++i) {
      int c   = i * 32 + lane;        // chunk id 0..255
      int rr  = c >> 4;               // tile row 0..15
      int cc  = c & 15;               // 16B chunk within row
      unsigned goff  = (unsigned)((((row0 + rr) * TLEN + t) * DIM + cc * 4) * 4);
      unsigned laddr = ldsb + (unsigned)((rr * TILE_STRIDE + cc * 4) * 4);
      asm volatile("global_load_async_to_lds_b128 %0, %1, %2"
                   :: "v"(laddr), "v"(goff), "s"(inputs) : "memory");
    }
  };

  prefetch(0, tb0);

#pragma unroll 1
  for (int t = 0; t < TLEN; ++t) {
    float*   cur  = (t & 1) ? tbuf1 : tbuf0;
    unsigned nxtb = (t & 1) ? tb0 : tb1;

    // Issue next tile's 8 DMAs first, then wait with threshold 8: only the
    // previous tile's DMAs must drain; the new ones stay in flight under
    // this whole iteration's compute.
    if (t + 1 < TLEN) {
      prefetch(t + 1, nxtb);
      asm volatile("s_wait_asynccnt 0x8" ::: "memory");
    } else {
      asm volatile("s_wait_asynccnt 0x0" ::: "memory");
    }

    // ---- x_proj tile: 16 chained f32 WMMAs per N-tile ----
    v8f cx0, cx1, cr0, cr1;
#pragma unroll
    for (int j = 0; j < 8; ++j) { cx0[j] = bin0; cx1[j] = bin1; cr0[j] = brc0; cr1[j] = brc1; }

    // A 16x4 layout: lanes 0-15 row=lane, K=4k+{0,1}; lanes 16-31 row=lane-16, K=4k+{2,3}
    const float* arow = cur + nlo * TILE_STRIDE + 2 * half;
#pragma unroll
    for (int k = 0; k < 16; ++k) {
      v2f a = *(const v2f*)(arow + 4 * k);
      cx0 = wmma_k4(a, bx0[k], cx0);
      cx1 = wmma_k4(a, bx1[k], cx1);
    }

    // ---- recurrent tile: h(16x12 in LDS) x rk ----
    const float* hrow = Hs + nlo * H_STRIDE + 2 * half;
#pragma unroll
    for (int kk = 0; kk < 3; ++kk) {
      v2f a = *(const v2f*)(hrow + 4 * kk);
      cr0 = wmma_k4(a, br0[kk], cr0);
      cr1 = wmma_k4(a, br1[kk], cr1);
    }

    // ---- stage C tiles to LDS for the (row, unit) gate relayout ----
    // C layout: VGPR j holds (M=j, N=lane) lanes0-15 and (M=j+8, N=lane-16) lanes16-31
    const int mbase = 8 * half;
#pragma unroll
    for (int j = 0; j < 8; ++j) {
      Sx[(mbase + j) * 32 + nlo]      = cx0[j];
      Sx[(mbase + j) * 32 + 16 + nlo] = cx1[j];
      Sr[(mbase + j) * 32 + nlo]      = cr0[j];
      Sr[(mbase + j) * 32 + 16 + nlo] = cr1[j];
    }

    // ---- gates: 160 (row,unit) entries, 5 per lane; same-wave LDS is in-order ----
#pragma unroll
    for (int e = 0; e < 5; ++e) {
      int idx = lane * 5 + e;
      int m = idx / UNITS, u = idx % UNITS;
      float xz = Sx[m * 32 + u],           rz = Sr[m * 32 + u];
      float xr = Sx[m * 32 + UNITS + u],   rr = Sr[m * 32 + UNITS + u];
      float xh = Sx[m * 32 + 2*UNITS + u], rh = Sr[m * 32 + 2*UNITS + u];
      float z  = fast_sigmoid(xz + rz);
      float r  = fast_sigmoid(xr + rr);
      float hh = fast_tanh(__builtin_fmaf(r, rh, xh));
      float hp = Hs[m * H_STRIDE + u];
      // z*hp + (1-z)*hh == hh + z*(hp - hh)
      Hs[m * H_STRIDE + u] = __builtin_fmaf(z, hp - hh, hh);
    }
  }

  // ---- write h_last (B, U) ----
#pragma unroll
  for (int e = 0; e < 5; ++e) {
    int idx = lane * 5 + e;
    int m = idx / UNITS, u = idx % UNITS;
    out[(row0 + m) * UNITS + u] = Hs[m * H_STRIDE + u];
  }
}

extern "C" void kernel_launch(void* const* d_in, const int* in_sizes, int n_in,
                              void* d_out, int out_size, void* d_ws, size_t ws_size,
                              hipStream_t stream) {
  const float* inputs = (const float*)d_in[0];   // (4096, 512, 64)
  const float* wk     = (const float*)d_in[1];   // (64, 30)
  const float* rk     = (const float*)d_in[2];   // (10, 30)
  const float* bias   = (const float*)d_in[3];   // (2, 30)
  float* out          = (float*)d_out;           // (4096, 10)

  const int tiles  = BATCH / ROWS_PER_WAVE;            // 256 waves
  const int blocks = tiles / WAVES_PER_BLOCK;          // 64 blocks
  gru_fused_kernel<<<blocks, WAVES_PER_BLOCK * 32, 0, stream>>>(inputs, wk, rk, bias, out);
}